// Vanilla_LSTM_15367392985737
// MI455X (gfx1250) — compile-verified
//
#include <hip/hip_runtime.h>

// ---------------------------------------------------------------------------
// Linear recurrence h_t = x_t @ W + h_{t-1} @ R on MI455X (gfx1250, wave32).
// Strategy: blocked (chunked) linear scan so nearly all FLOPs run through
// V_WMMA_F32_16X16X4_F32 (the only true-fp32 WMMA on CDNA5), preserving
// reference fp32 numerics.
//   K1: xW = x @ W            (65536x512x512 WMMA GEMM, into d_out in place)
//   K2: R^2..R^64             (6 squaring GEMMs in d_ws)
//   K3: pass A  y_j = xw_j + y_{j-1} R   per (batch,chunk) row, carries -> ws
//   K4: carry scan E_k = c_k + E_{k-1} R^64   (32 steps)
//   K5: pass B  z_j = z_{j-1} R (seeded E_{k-1}); h = y + z  (in d_out)
// ---------------------------------------------------------------------------

typedef float v2f __attribute__((ext_vector_type(2)));
typedef float v8f __attribute__((ext_vector_type(8)));

// 8-arg form: (neg_a, A, neg_b, B, c_mod, C, reuse_a, reuse_b)
#define WMMA_F32(a, b, c) \
  __builtin_amdgcn_wmma_f32_16x16x4_f32(false, (a), false, (b), (short)0, (c), false, false)

namespace {
constexpr int B_ = 32;
constexpr int T_ = 2048;
constexpr int U_ = 512;          // == D
constexpr int L_ = 64;           // chunk length
constexpr int C_ = T_ / L_;      // 32 chunks
constexpr int ROWS_ = B_ * C_;   // 1024 independent scan rows
}

// ---------------------------------------------------------------------------
// Generic fp32 GEMM, K = N = 512, 16 rows per block, 8 waves x 64 cols each.
// A strip (16x512, 32KB) staged in LDS so x is read once from HBM; B streams
// from L2 (W / R are 1MB each and stay resident in the 192MB L2).
// ---------------------------------------------------------------------------
__global__ __launch_bounds__(256) void lr_gemm_f32_k512(
    const float* __restrict__ A, const float* __restrict__ Bm,
    float* __restrict__ Cout) {
  __shared__ float As[16 * 512];
  const int tid  = threadIdx.x;
  const int wave = tid >> 5;
  const int lane = tid & 31;
  const int hi   = lane >> 4;   // 0: K even pair, 1: K odd pair (per frag)
  const int lo   = lane & 15;
  const int m0   = blockIdx.x * 16;

  for (int e = tid; e < 16 * 512; e += 256) {
    int r = e >> 9, c = e & 511;
    As[e] = A[(size_t)(m0 + r) * 512 + c];
  }
  __syncthreads();

  const int nBase = wave * 64;  // 4 n-tiles of 16 per wave
  v8f acc[4] = {};
  for (int kk = 0; kk < 512; kk += 4) {
    v2f a = *(const v2f*)&As[lo * 512 + kk + 2 * hi];  // A[M=lo][k, k+1]
#pragma unroll
    for (int nt = 0; nt < 4; ++nt) {
      int col = nBase + nt * 16 + lo;
      v2f b;
      b.x = Bm[(size_t)(kk + 2 * hi) * 512 + col];      // B[k][N=col]
      b.y = Bm[(size_t)(kk + 2 * hi + 1) * 512 + col];  // B[k+1][N=col]
      acc[nt] = WMMA_F32(a, b, acc[nt]);
    }
  }
#pragma unroll
  for (int nt = 0; nt < 4; ++nt) {
    int col = nBase + nt * 16 + lo;
#pragma unroll
    for (int jv = 0; jv < 8; ++jv) {
      int row = jv + 8 * hi;  // C layout: VGPR jv -> M = jv (lo half) / jv+8
      Cout[(size_t)(m0 + row) * 512 + col] = acc[nt][jv];
    }
  }
}

// ---------------------------------------------------------------------------
// Recurrence over LL steps for 16 independent rows (H state 16x512 in LDS,
// ping-pong + one barrier per step).  Row r -> (b = r/CC, k = r%CC),
// io element index = (b*CC*LL + k*LL + j)*512 + u.
//  MODE 0 (pass A):  H=0;        acc = io(j); acc += H@R; io(j)=acc; H=acc;
//                    final H -> carry[r].
//  MODE 1 (carry):   H=h0[b];    acc = io(j); acc += H@R; io(j)=acc; H=acc.
//  MODE 2 (pass B):  H=E_{k-1} (h0 if k==0); z = H@R; io(j) += z; H=z.
// ---------------------------------------------------------------------------
template <int MODE>
__global__ __launch_bounds__(256) void lr_recur(
    float* __restrict__ io, const float* __restrict__ Rm,
    float* __restrict__ carry, const float* __restrict__ h0,
    int CC, int LL) {
  __shared__ float Hs[2][16 * 512];  // 64KB ping-pong state
  __shared__ int rowBase[16];
  const int tid  = threadIdx.x;
  const int wave = tid >> 5;
  const int lane = tid & 31;
  const int hi   = lane >> 4;
  const int lo   = lane & 15;
  const int row0 = blockIdx.x * 16;

  if (tid < 16) {
    int r = row0 + tid;
    int b = r / CC, k = r % CC;
    rowBase[tid] = b * CC * LL + k * LL;
  }
  // Initialize H state.
  for (int e = tid; e < 16 * 512; e += 256) {
    int m = e >> 9, c = e & 511;
    float v = 0.f;
    if (MODE == 1) {
      v = h0[(row0 + m) * 512 + c];  // CC==1 -> row index is batch index
    } else if (MODE == 2) {
      int r = row0 + m;
      int b = r / CC, k = r % CC;
      v = (k == 0) ? h0[b * 512 + c] : carry[(size_t)(r - 1) * 512 + c];
    }
    Hs[0][e] = v;
  }
  __syncthreads();

  const int nBase = wave * 64;
  int cur = 0;
  for (int j = 0; j < LL; ++j) {
    v8f acc[4] = {};
    if (MODE != 2) {  // accumulate on top of the input tile
#pragma unroll
      for (int nt = 0; nt < 4; ++nt) {
        int col = nBase + nt * 16 + lo;
#pragma unroll
        for (int jv = 0; jv < 8; ++jv) {
          int row = jv + 8 * hi;
          acc[nt][jv] = io[(size_t)(rowBase[row] + j) * 512 + col];
        }
      }
    }
    for (int kk = 0; kk < 512; kk += 4) {
      v2f a = *(const v2f*)&Hs[cur][lo * 512 + kk + 2 * hi];
#pragma unroll
      for (int nt = 0; nt < 4; ++nt) {
        int col = nBase + nt * 16 + lo;
        v2f b;
        b.x = Rm[(size_t)(kk + 2 * hi) * 512 + col];
        b.y = Rm[(size_t)(kk + 2 * hi + 1) * 512 + col];
        acc[nt] = WMMA_F32(a, b, acc[nt]);
      }
    }
#pragma unroll
    for (int nt = 0; nt < 4; ++nt) {
      int col = nBase + nt * 16 + lo;
#pragma unroll
      for (int jv = 0; jv < 8; ++jv) {
        int row = jv + 8 * hi;
        size_t adr = (size_t)(rowBase[row] + j) * 512 + col;
        float z = acc[nt][jv];
        if (MODE == 2) io[adr] = io[adr] + z;
        else           io[adr] = z;
        Hs[cur ^ 1][row * 512 + col] = z;  // next-step state
      }
    }
    __syncthreads();
    cur ^= 1;
  }
  if (MODE == 0) {  // chunk-end carry c_r = y_{L-1}
    for (int e = tid; e < 16 * 512; e += 256) {
      int m = e >> 9, c = e & 511;
      carry[(size_t)(row0 + m) * 512 + c] = Hs[cur][e];
    }
  }
}

extern "C" void kernel_launch(void* const* d_in, const int* in_sizes, int n_in,
                              void* d_out, int out_size, void* d_ws, size_t ws_size,
                              hipStream_t stream) {
  (void)in_sizes; (void)n_in; (void)out_size; (void)ws_size;
  const float* x  = (const float*)d_in[0];  // [B,T,D]
  const float* W  = (const float*)d_in[1];  // [D,U]
  const float* R  = (const float*)d_in[2];  // [U,U]
  const float* h0 = (const float*)d_in[3];  // [B,U]
  float* out = (float*)d_out;               // [B,T,U], used in-place

  // d_ws layout: pow0 (1MB) | pow1 (1MB) | carries (B*C*U*4 = 2MB)
  float* pow0  = (float*)d_ws;
  float* pow1  = pow0 + 512 * 512;
  float* carry = pow1 + 512 * 512;

  // 1) xW -> out  (M = B*T = 65536)
  lr_gemm_f32_k512<<<(B_ * T_) / 16, 256, 0, stream>>>(x, W, out);

  // 2) R^64 by repeated squaring (6 GEMMs of 512^3, final result in pow1)
  lr_gemm_f32_k512<<<512 / 16, 256, 0, stream>>>(R,    R,    pow0);  // R^2
  lr_gemm_f32_k512<<<512 / 16, 256, 0, stream>>>(pow0, pow0, pow1);  // R^4
  lr_gemm_f32_k512<<<512 / 16, 256, 0, stream>>>(pow1, pow1, pow0);  // R^8
  lr_gemm_f32_k512<<<512 / 16, 256, 0, stream>>>(pow0, pow0, pow1);  // R^16
  lr_gemm_f32_k512<<<512 / 16, 256, 0, stream>>>(pow1, pow1, pow0);  // R^32
  lr_gemm_f32_k512<<<512 / 16, 256, 0, stream>>>(pow0, pow0, pow1);  // R^64

  // 3) pass A: local scans over 1024 rows (64 blocks), carries -> ws
  lr_recur<0><<<ROWS_ / 16, 256, 0, stream>>>(out, R, carry, h0, C_, L_);

  // 4) carry scan: E_k = c_k + E_{k-1} @ R^64 in-place on carry (32 steps)
  lr_recur<1><<<B_ / 16, 256, 0, stream>>>(carry, pow1, nullptr, h0, 1, C_);

  // 5) pass B: z_j = z_{j-1} @ R seeded with E_{k-1}; h = y + z in-place
  lr_recur<2><<<ROWS_ / 16, 256, 0, stream>>>(out, R, carry, h0, C_, L_);
}